// RefinementLoss_72078141161695
// MI455X (gfx1250) — compile-verified
//
#include <hip/hip_runtime.h>

#define DEVFN __device__ __forceinline__

constexpr int Bc = 4, Fc = 512, Nc = 1024, Vc = 4;
constexpr int NBLK = Bc * Fc; // 2048 (b,f) blocks
constexpr float IMGc = 224.0f;

typedef __attribute__((ext_vector_type(2))) float v2f;
typedef __attribute__((ext_vector_type(8))) float v8f;

// ---- monotone float <-> uint encoding so we can use integer atomicMin/Max ----
DEVFN unsigned encF(float f) {
  unsigned u = __float_as_uint(f);
  return (u & 0x80000000u) ? ~u : (u | 0x80000000u);
}
DEVFN float decF(unsigned u) {
  return (u & 0x80000000u) ? __uint_as_float(u & 0x7FFFFFFFu)
                           : __uint_as_float(~u);
}

// ---- wave32 reductions ----
DEVFN float waveSum(float v) {
#pragma unroll
  for (int o = 16; o > 0; o >>= 1) v += __shfl_down(v, o, 32);
  return v;
}
DEVFN unsigned waveMinU(unsigned v) {
#pragma unroll
  for (int o = 16; o > 0; o >>= 1) {
    unsigned t = __shfl_down(v, o, 32);
    v = t < v ? t : v;
  }
  return v;
}
DEVFN unsigned waveMaxU(unsigned v) {
#pragma unroll
  for (int o = 16; o > 0; o >>= 1) {
    unsigned t = __shfl_down(v, o, 32);
    v = t > v ? t : v;
  }
  return v;
}

// ws layout (32-bit words in d_ws):
//  floats: [0] sum_diff2  [1] visible-point count  [14] sum_acc2  [15] sum_ident
//  uints : [2..4] gt-min enc  [5..7] gt-max enc  [8..10] pred-min enc [11..13] pred-max enc
__global__ void init_ws(float* wsF, unsigned* wsU) {
  int t = threadIdx.x;
  if (t == 0) wsF[0] = 0.f;
  else if (t == 1) wsF[1] = 0.f;
  else if (t == 2) wsF[14] = 0.f;
  else if (t == 3) wsF[15] = 0.f;
  else if (t < 16) {
    int q = t - 4;
    bool isMin = (q < 3) || (q >= 6 && q < 9);
    wsU[2 + q] = isMin ? 0xFFFFFFFFu : 0u;
  }
}

__global__ __launch_bounds__(256) void loss_main(
    const float* __restrict__ pred, const float* __restrict__ gt,
    const int* __restrict__ vis, const float* __restrict__ P,
    const float* __restrict__ tracks, float* wsF, unsigned* wsU) {
  const int bf = blockIdx.x;
  const int b = bf / Fc, f = bf % Fc;
  const int tid = threadIdx.x;
  const int lane = tid & 31, wave = tid >> 5;

  const size_t frameBase = ((size_t)b * Fc + f) * (size_t)Nc;
  const float* predF = pred + frameBase * 3;
  const float* gtF = gt + frameBase * 3;
  const int* visF = vis + frameBase;

  float sdiff2 = 0.f, scnt = 0.f, sacc2 = 0.f, sid = 0.f;
  unsigned gmin0 = 0xFFFFFFFFu, gmin1 = 0xFFFFFFFFu, gmin2 = 0xFFFFFFFFu;
  unsigned gmax0 = 0u, gmax1 = 0u, gmax2 = 0u;
  unsigned pmn0 = 0xFFFFFFFFu, pmn1 = 0xFFFFFFFFu, pmn2 = 0xFFFFFFFFu;
  unsigned pmx0 = 0u, pmx1 = 0u, pmx2 = 0u;

  const bool doTemp = (f >= 1) && (f <= Fc - 2); // uniform per block
  const float* predM = predF - (size_t)Nc * 3;
  const float* predP = predF + (size_t)Nc * 3;

  // ---------- elementwise: recon + temporal + min/max (branchless) ----------
  for (int n = tid; n < Nc; n += 256) {
    float px = predF[n * 3 + 0], py = predF[n * 3 + 1], pz = predF[n * 3 + 2];
    unsigned ex = encF(px), ey = encF(py), ez = encF(pz);
    pmn0 = ex < pmn0 ? ex : pmn0; pmx0 = ex > pmx0 ? ex : pmx0;
    pmn1 = ey < pmn1 ? ey : pmn1; pmx1 = ey > pmx1 ? ey : pmx1;
    pmn2 = ez < pmn2 ? ez : pmn2; pmx2 = ez > pmx2 ? ez : pmx2;

    float gx = gtF[n * 3 + 0], gy = gtF[n * 3 + 1], gz = gtF[n * 3 + 2];
    const bool vb = visF[n] != 0;
    const float vm = vb ? 1.0f : 0.0f;
    float dx = px - gx, dy = py - gy, dz = pz - gz;
    sdiff2 += vm * (dx * dx + dy * dy + dz * dz);
    scnt += vm;
    unsigned hx = encF(gx), hy = encF(gy), hz = encF(gz);
    unsigned lx = vb ? hx : 0xFFFFFFFFu, ly = vb ? hy : 0xFFFFFFFFu,
             lz = vb ? hz : 0xFFFFFFFFu;
    unsigned ux = vb ? hx : 0u, uy = vb ? hy : 0u, uz = vb ? hz : 0u;
    gmin0 = lx < gmin0 ? lx : gmin0; gmax0 = ux > gmax0 ? ux : gmax0;
    gmin1 = ly < gmin1 ? ly : gmin1; gmax1 = uy > gmax1 ? uy : gmax1;
    gmin2 = lz < gmin2 ? lz : gmin2; gmax2 = uz > gmax2 ? uz : gmax2;

    if (doTemp) { // acc = pred[f+1] - 2*pred[f] + pred[f-1]  (L2-resident reads)
      float ax = predP[n * 3 + 0] - 2.0f * px + predM[n * 3 + 0];
      float ay = predP[n * 3 + 1] - 2.0f * py + predM[n * 3 + 1];
      float az = predP[n * 3 + 2] - 2.0f * pz + predM[n * 3 + 2];
      sacc2 += ax * ax + ay * ay + az * az;
    }
  }

  // ---------- identity loss via V_WMMA_F32_16X16X4_F32 ----------
  // D(16x16) = A(16 points x 4 homo) * B(4 x 12 projection rows, padded to 16)
  const int n16 = lane & 15;
  const bool hi = lane >= 16;
  const int vv = n16 / 3, jj = n16 - vv * 3;
  const bool valid = (n16 < 12) && (jj < 2);
  const float vmask = valid ? 1.0f : 0.0f;
  // clamped (v,c) for invalid lanes: lands on an adjacent valid lane's cache
  // line, so no extra lines are fetched; contribution is masked to zero.
  const int vvc = (n16 < 12) ? vv : 3;
  const int jjc = valid ? jj : 1;

  v2f bmat;
  bmat.x = 0.f; bmat.y = 0.f;
  if (n16 < 12) {
    // B[k][n16] = P[vv, b, jj, k]; lanes 0-15 hold K=0,1; lanes 16-31 hold K=2,3
    const float* Pr = P + (((size_t)vv * Bc + b) * 3 + jj) * 4;
    int k0 = hi ? 2 : 0;
    bmat.x = Pr[k0];
    bmat.y = Pr[k0 + 1];
  }
  const int srcZ = (hi ? 16 : 0) + (n16 < 12 ? vv * 3 + 2 : n16);
  const float* trackRow =
      tracks + ((((size_t)vvc * Bc + b) * Fc + f) * Nc) * 2 + jjc;

  for (int t = 0; t < 8; ++t) { // 8 tiles of 16 points per wave -> 128 points
    int base = wave * 128 + t * 16;
    int pt = base + n16;
    const float* pp = predF + (size_t)pt * 3;
    float x = pp[0], y = pp[1], z = pp[2];
    v2f amat; // A row = point n16; lanes 0-15: K=0,1 (x,y); lanes 16-31: K=2,3 (z,1)
    amat.x = hi ? z : x;
    amat.y = hi ? 1.0f : y;
    v8f d = {};
    d = __builtin_amdgcn_wmma_f32_16x16x4_f32(
        /*neg_a=*/false, amat, /*neg_b=*/false, bmat,
        /*c_mod=*/(short)0, d, /*reuse_a=*/false, /*reuse_b=*/false);
#pragma unroll
    for (int r = 0; r < 8; ++r) {
      float val = d[r];                  // D[M][n16], M = r + (hi?8:0)
      float zv = __shfl(val, srcZ, 32);  // denominator from column 3v+2
      float p2 = val * __builtin_amdgcn_rcpf(zv + 1e-10f); // v_rcp + v_mul
      int ptM = base + r + (hi ? 8 : 0);
      float tr = trackRow[(size_t)ptM * 2];
      float dd = p2 - tr;
      sid += vmask * (dd * dd);          // branchless masked accumulate
    }
  }

  // ---------- block reduction + global atomics ----------
  sdiff2 = waveSum(sdiff2); scnt = waveSum(scnt);
  sacc2 = waveSum(sacc2);   sid = waveSum(sid);
  gmin0 = waveMinU(gmin0); gmin1 = waveMinU(gmin1); gmin2 = waveMinU(gmin2);
  gmax0 = waveMaxU(gmax0); gmax1 = waveMaxU(gmax1); gmax2 = waveMaxU(gmax2);
  pmn0 = waveMinU(pmn0); pmn1 = waveMinU(pmn1); pmn2 = waveMinU(pmn2);
  pmx0 = waveMaxU(pmx0); pmx1 = waveMaxU(pmx1); pmx2 = waveMaxU(pmx2);

  __shared__ float shF[4][8];
  __shared__ unsigned shU[12][8];
  if (lane == 0) {
    shF[0][wave] = sdiff2; shF[1][wave] = scnt;
    shF[2][wave] = sacc2;  shF[3][wave] = sid;
    shU[0][wave] = gmin0; shU[1][wave] = gmin1; shU[2][wave] = gmin2;
    shU[3][wave] = gmax0; shU[4][wave] = gmax1; shU[5][wave] = gmax2;
    shU[6][wave] = pmn0;  shU[7][wave] = pmn1;  shU[8][wave] = pmn2;
    shU[9][wave] = pmx0;  shU[10][wave] = pmx1; shU[11][wave] = pmx2;
  }
  __syncthreads();
  if (tid < 4) {
    const int fmap[4] = {0, 1, 14, 15};
    float s = 0.f;
#pragma unroll
    for (int i = 0; i < 8; ++i) s += shF[tid][i];
    atomicAdd(&wsF[fmap[tid]], s);
  } else if (tid < 16) {
    int q = tid - 4;
    bool isMin = (q < 3) || (q >= 6 && q < 9);
    unsigned r = isMin ? 0xFFFFFFFFu : 0u;
#pragma unroll
    for (int i = 0; i < 8; ++i) {
      unsigned t2 = shU[q][i];
      r = isMin ? (t2 < r ? t2 : r) : (t2 > r ? t2 : r);
    }
    if (isMin) atomicMin(&wsU[2 + q], r);
    else       atomicMax(&wsU[2 + q], r);
  }
}

__global__ void finalize_k(const float* wsF, const unsigned* wsU, float* out) {
  if (threadIdx.x != 0 || blockIdx.x != 0) return;
  // recon
  float cnt = wsF[1] * 3.0f;
  float rg = -3.4e38f;
  for (int c = 0; c < 3; ++c) {
    float d = decF(wsU[5 + c]) - decF(wsU[2 + c]);
    rg = fmaxf(rg, d);
  }
  float scaleR = rg + 1e-6f;
  float recon = wsF[0] / fmaxf(cnt, 1.0f) / (scaleR * scaleR);
  // identity: mean over [V,B,F,N,2] of diff^2 / IMG
  float ident = wsF[15] / (IMGc * (float)Vc * (float)Bc * (float)Fc * (float)Nc * 2.0f);
  // temporal
  float tl = wsF[14] / ((float)Bc * (float)(Fc - 2) * (float)Nc);
  float rp = -3.4e38f;
  for (int c = 0; c < 3; ++c) {
    float d = decF(wsU[11 + c]) - decF(wsU[8 + c]);
    rp = fmaxf(rp, d);
  }
  float scaleT = rp + 1e-6f;
  float temp = tl / (scaleT * scaleT);
  out[0] = recon + ident + 0.5f * temp;
  out[1] = recon;
  out[2] = ident;
  out[3] = temp;
}

extern "C" void kernel_launch(void* const* d_in, const int* in_sizes, int n_in,
                              void* d_out, int out_size, void* d_ws, size_t ws_size,
                              hipStream_t stream) {
  (void)in_sizes; (void)n_in; (void)out_size; (void)ws_size;
  const float* pred = (const float*)d_in[0];
  const float* gt = (const float*)d_in[1];
  const int* vis = (const int*)d_in[2];
  const float* P = (const float*)d_in[3];
  const float* tracks = (const float*)d_in[4];
  float* wsF = (float*)d_ws;
  unsigned* wsU = (unsigned*)d_ws;
  float* out = (float*)d_out;

  init_ws<<<1, 32, 0, stream>>>(wsF, wsU);
  loss_main<<<NBLK, 256, 0, stream>>>(pred, gt, vis, P, tracks, wsF, wsU);
  finalize_k<<<1, 32, 0, stream>>>(wsF, wsU, out);
}